// DistMult_1700807049273
// MI455X (gfx1250) — compile-verified
//
#include <hip/hip_runtime.h>

// DistMult bilinear scoring: scores[e] = left[e] @ W[r_id[e]] @ right[e]
// E = 4,194,304 edges, DIM = 32, NUM_REL = 8, all fp32.
//
// Strategy (MI455X / gfx1250):
//  - memory-bound target (~1.06 GB -> ~45us @ 23.3 TB/s): one pass over
//    left/right, 32x32 matvec on the matrix pipe (V_WMMA_F32_16X16X4_F32)
//  - per-block LDS binning by relation so each 16-edge WMMA tile shares one W
//  - W stored K-interleaved in LDS: B fragment = single conflict-free
//    ds_load_b64 into an even-aligned VGPR pair (no repacking movs)

#define DIM            32
#define NUM_REL        8
#define CHUNK          2048        // edges per block
#define BLOCK_THREADS  256         // 8 wave32
#define WAVES_PER_BLK  8
#define LWPAD          36          // padded LW tile row stride (dwords, 16B aligned)

typedef __attribute__((ext_vector_type(2))) float v2f;
typedef __attribute__((ext_vector_type(4))) float v4f;
typedef __attribute__((ext_vector_type(8))) float v8f;

__global__ __launch_bounds__(BLOCK_THREADS)
void distmult_wmma_kernel(const float* __restrict__ left,
                          const float* __restrict__ right,
                          const int*   __restrict__ rid,
                          const float* __restrict__ W,
                          float* __restrict__ out,
                          int E)
{
    // K-interleaved W: element (r, row kr, col n) at ((r*8 + kr/4)*32 + n)*4 + (kr&3)
    __shared__ float          sW4[NUM_REL * DIM * DIM];          // 32 KB, no pad
    __shared__ unsigned short sBucket[NUM_REL][CHUNK];           // 32 KB
    __shared__ unsigned int   sCount[NUM_REL];
    __shared__ unsigned int   sTile[NUM_REL + 1];
    __shared__ float          sLW[WAVES_PER_BLK][16 * LWPAD];    // ~18 KB

    const int tid        = threadIdx.x;
    const int blockStart = blockIdx.x * CHUNK;

    if (tid < NUM_REL) sCount[tid] = 0u;

    // Load W (8 x 32 x 32 fp32) into K-interleaved LDS; reused by all tiles.
    for (int i = tid; i < NUM_REL * DIM * DIM; i += BLOCK_THREADS) {
        int r  = i >> 10;
        int kr = (i >> 5) & 31;
        int n  = i & 31;
        int k  = kr >> 2;        // K-step (0..7)
        int c  = kr & 3;         // row within K-step
        sW4[((r * 8 + k) * 32 + n) * 4 + c] = W[i];
    }
    __syncthreads();

    // ---- Phase 1: bucket this block's edges by relation id (LDS atomics) ----
#pragma unroll
    for (int j = 0; j < CHUNK / BLOCK_THREADS; ++j) {
        int li = tid + j * BLOCK_THREADS;          // local edge index
        int gi = blockStart + li;
        if (gi < E) {
            int r = rid[gi];
            unsigned int slot = atomicAdd(&sCount[r], 1u);
            sBucket[r][slot] = (unsigned short)li;
        }
    }
    __syncthreads();

    if (tid == 0) {
        unsigned int acc = 0u;
        for (int r = 0; r < NUM_REL; ++r) {
            sTile[r] = acc;
            acc += (sCount[r] + 15u) >> 4;         // 16-edge tiles per relation
        }
        sTile[NUM_REL] = acc;
    }
    __syncthreads();

    const int wave     = tid >> 5;
    const int lane     = tid & 31;
    const int halfLane = lane & 15;
    const int hi       = lane >> 4;     // 0: lanes 0-15, 1: lanes 16-31
    const int koff     = hi << 1;       // A-fragment K offset for this half
    const unsigned int numTiles = sTile[NUM_REL];

    // ---- Phase 2: each wave consumes relation-uniform 16-edge tiles ----
    for (unsigned int t = wave; t < numTiles; t += WAVES_PER_BLK) {
        int r = 0;
        while (r < NUM_REL - 1 && t >= sTile[r + 1]) ++r;
        const unsigned int cnt      = sCount[r];
        const unsigned int rowStart = (t - sTile[r]) << 4;
        const unsigned int rows     = (cnt - rowStart < 16u) ? (cnt - rowStart) : 16u;

        // Per ISA A-layout: lanes l and l+16 both own row M=l.
        const unsigned int mrow = (unsigned int)halfLane;
        const unsigned int sel  = (mrow < rows) ? mrow : 0u;   // clamp pad rows
        const int  eLocal = sBucket[r][rowStart + sel];
        const long long gEdge = (long long)(blockStart + eLocal);
        const float* lrow = left + gEdge * DIM;

        // A fragments for all 8 K-steps: VGPR0/1 = K=4k+koff, 4k+koff+1
        // -> one contiguous 8B load per step (whole row = one 128B line).
        v2f a[8];
#pragma unroll
        for (int k = 0; k < 8; ++k) {
            a[k] = *(const v2f*)(lrow + 4 * k + koff);
        }

        const float* wr = &sW4[(r * 8) * 32 * 4];

        // LW(16x32) = L(16x32) @ W_r(32x32): 2 column blocks x 8 K-steps
#pragma unroll
        for (int cb = 0; cb < 2; ++cb) {
            v8f c = {};
#pragma unroll
            for (int k = 0; k < 8; ++k) {
                // B fragment: rows K=4k+koff, 4k+koff+1 at col n = cb*16+halfLane,
                // adjacent in the K-interleaved layout -> one ds_load_b64.
                v2f b = *(const v2f*)(wr + (k * 32 + cb * 16 + halfLane) * 4 + koff);
                c = __builtin_amdgcn_wmma_f32_16x16x4_f32(
                        /*neg_a=*/false, a[k], /*neg_b=*/false, b,
                        /*c_mod=*/(short)0, c, /*reuse_a=*/false, /*reuse_b=*/false);
            }
            // Spill C fragment to LDS in natural (m,n) order:
            // VGPR v, lane -> (M = v + 8*hi, N = halfLane + 16*cb)
            float* dst = &sLW[wave][0];
#pragma unroll
            for (int v = 0; v < 8; ++v) {
                int m = v + 8 * hi;
                dst[m * LWPAD + cb * 16 + halfLane] = c[v];
            }
        }

        // Dot with right row: lane pair (m, m+16) splits the 32 columns.
        const float* rrow  = right + gEdge * DIM + hi * 16;
        const float* lwrow = &sLW[wave][(int)mrow * LWPAD + hi * 16];
        float partial = 0.f;
#pragma unroll
        for (int q = 0; q < 4; ++q) {
            v4f lw = *(const v4f*)(lwrow + 4 * q);
            v4f rv = *(const v4f*)(rrow + 4 * q);
            partial += lw.x * rv.x + lw.y * rv.y + lw.z * rv.z + lw.w * rv.w;
        }
        float score = partial + __shfl_xor(partial, 16, 32);
        if (hi == 0 && mrow < rows) {
            out[gEdge] = score;
        }
    }
}

extern "C" void kernel_launch(void* const* d_in, const int* in_sizes, int n_in,
                              void* d_out, int out_size, void* d_ws, size_t ws_size,
                              hipStream_t stream) {
    const float* left  = (const float*)d_in[0];   // [E, 32] fp32
    const float* right = (const float*)d_in[1];   // [E, 32] fp32
    const int*   rid   = (const int*)d_in[2];     // [E] int32
    const float* W     = (const float*)d_in[3];   // [8, 32, 32] fp32
    float*       out   = (float*)d_out;           // [E] fp32

    const int E = in_sizes[2];
    const int blocks = (E + CHUNK - 1) / CHUNK;
    hipLaunchKernelGGL(distmult_wmma_kernel, dim3(blocks), dim3(BLOCK_THREADS),
                       0, stream, left, right, rid, W, out, E);
}